// Earth4D_77000173683292
// MI455X (gfx1250) — compile-verified
//
#include <hip/hip_runtime.h>
#include <math.h>

// Earth4D multi-grid hash encoding, MI455X (gfx1250, wave32).
// One thread = one point. Grid-outer / level-inner loops keep all resident
// waves walking the same 4MB table slab together -> gathers hit the 192MB L2.
// Hashed-level loop is unrolled 2x so ~16 global_load_b64 gathers are in
// flight per wave (next level's addresses don't depend on this level's data).

#define N_LEVELS   24
#define LOG2_T     19
#define TSIZE      (1u << LOG2_T)
#define TMASK      (TSIZE - 1u)
#define PRIME1     2654435761u
#define PRIME2     805459861u

typedef float v2f __attribute__((ext_vector_type(2)));
typedef float v4f __attribute__((ext_vector_type(4)));

__device__ __forceinline__ float clamp01(float v) {
    return fminf(fmaxf(v, 0.0f), 1.0f);
}

// One level of one grid: 8 corner gathers + trilinear blend -> float2.
// DENSE=true uses lattice indexing (levels 0-1, R compile-time foldable);
// DENSE=false uses the NGP spatial hash.
template <bool DENSE>
__device__ __forceinline__ void encode_level(int R,
                                             float u0, float u1, float u2,
                                             const float* __restrict__ lt,
                                             float* __restrict__ outPos) {
    const float Rf  = (float)R;
    const float Rm1 = Rf - 1.0f;   // matches f32(R-1) incl. rounding at 2^28

    float pos0 = u0 * Rf, pos1 = u1 * Rf, pos2 = u2 * Rf;
    float f0 = fminf(floorf(pos0), Rm1);
    float f1 = fminf(floorf(pos1), Rm1);
    float f2 = fminf(floorf(pos2), Rm1);
    float w0 = pos0 - f0, w1 = pos1 - f1, w2 = pos2 - f2;

    unsigned p0 = (unsigned)(int)f0;
    unsigned p1 = (unsigned)(int)f1;
    unsigned p2 = (unsigned)(int)f2;

    unsigned i0, i1, i2, i3, i4, i5, i6, i7;
    if (DENSE) {
        const unsigned R1  = (unsigned)R + 1u;
        const unsigned R1s = R1 * R1;
        unsigned x0 = p0 * R1s, x1 = x0 + R1s;     // corner bit0 -> dim0
        unsigned y0 = p1 * R1,  y1 = y0 + R1;      // corner bit1 -> dim1
        unsigned z0 = p2,       z1 = p2 + 1u;      // corner bit2 -> dim2
        i0 = x0 + y0 + z0;  i1 = x1 + y0 + z0;
        i2 = x0 + y1 + z0;  i3 = x1 + y1 + z0;
        i4 = x0 + y0 + z1;  i5 = x1 + y0 + z1;
        i6 = x0 + y1 + z1;  i7 = x1 + y1 + z1;
    } else {
        unsigned x0 = p0,          x1 = p0 + 1u;
        unsigned y0 = p1 * PRIME1, y1 = y0 + PRIME1;
        unsigned z0 = p2 * PRIME2, z1 = z0 + PRIME2;
        i0 = (x0 ^ y0 ^ z0) & TMASK;  i1 = (x1 ^ y0 ^ z0) & TMASK;
        i2 = (x0 ^ y1 ^ z0) & TMASK;  i3 = (x1 ^ y1 ^ z0) & TMASK;
        i4 = (x0 ^ y0 ^ z1) & TMASK;  i5 = (x1 ^ y0 ^ z1) & TMASK;
        i6 = (x0 ^ y1 ^ z1) & TMASK;  i7 = (x1 ^ y1 ^ z1) & TMASK;
    }

    // Issue all 8 gathers (global_load_b64 each) before consuming.
    const v2f* tp = (const v2f*)lt;
    v2f g0 = tp[i0]; v2f g1 = tp[i1]; v2f g2 = tp[i2]; v2f g3 = tp[i3];
    v2f g4 = tp[i4]; v2f g5 = tp[i5]; v2f g6 = tp[i6]; v2f g7 = tp[i7];

    // Trilinear weights (corner bit set -> w, clear -> 1-w).
    float o0 = 1.0f - w0, o1 = 1.0f - w1, o2 = 1.0f - w2;
    float m00 = o1 * o2, m01 = w1 * o2, m10 = o1 * w2, m11 = w1 * w2;
    float wt0 = o0 * m00, wt1 = w0 * m00;
    float wt2 = o0 * m01, wt3 = w0 * m01;
    float wt4 = o0 * m10, wt5 = w0 * m10;
    float wt6 = o0 * m11, wt7 = w0 * m11;

    float a = wt0 * g0.x; float b = wt0 * g0.y;
    a = fmaf(wt1, g1.x, a);  b = fmaf(wt1, g1.y, b);
    a = fmaf(wt2, g2.x, a);  b = fmaf(wt2, g2.y, b);
    a = fmaf(wt3, g3.x, a);  b = fmaf(wt3, g3.y, b);
    a = fmaf(wt4, g4.x, a);  b = fmaf(wt4, g4.y, b);
    a = fmaf(wt5, g5.x, a);  b = fmaf(wt5, g5.y, b);
    a = fmaf(wt6, g6.x, a);  b = fmaf(wt6, g6.y, b);
    a = fmaf(wt7, g7.x, a);  b = fmaf(wt7, g7.y, b);

    v2f r; r.x = a; r.y = b;
    *(v2f*)outPos = r;                         // global_store_b64, 8B aligned
}

// 24 levels of one grid; results go to outChunk[level*2 .. level*2+1].
__device__ __forceinline__ void encode_grid(float u0, float u1, float u2,
                                            const float* __restrict__ table,
                                            float* __restrict__ outChunk) {
    // Levels 0-1: dense lattice indexing, R folds to literals 32/64.
    encode_level<true>(32, u0, u1, u2, table,                     outChunk);
    encode_level<true>(64, u0, u1, u2, table + (size_t)TSIZE * 2, outChunk + 2);

    // Levels 2-23: hashed. Unroll 2x -> two levels' gathers in flight.
    const float* lt = table + (size_t)2 * TSIZE * 2;
#pragma unroll 2
    for (int level = 2; level < N_LEVELS; ++level) {
        encode_level<false>(32 << level, u0, u1, u2, lt, outChunk + level * 2);
        lt += (size_t)TSIZE * 2;
    }
}

__global__ void __launch_bounds__(256)
earth4d_kernel(const v4f* __restrict__ coords,
               const float* __restrict__ t_spatial,
               const float* __restrict__ t_xyt,
               const float* __restrict__ t_yzt,
               const float* __restrict__ t_xzt,
               float* __restrict__ out, int n) {
    int i = blockIdx.x * blockDim.x + threadIdx.x;
    if (i >= n) return;

    v4f c = __builtin_nontemporal_load(coords + i);   // streamed once

    // WGS84 -> ECEF (once per point; amortized over 96 level passes)
    const float DEG2RAD = 0.017453292519943295f;
    const float A_WGS   = 6378137.0f;
    const float E2      = 6.6943799901413165e-3f;     // f*(2-f), f = 1/298.257223563
    float lat = c.x * DEG2RAD;
    float lon = c.y * DEG2RAD;
    float elev = c.z;
    float t    = c.w;
    float sl = sinf(lat), cl = cosf(lat);
    float so = sinf(lon), co = cosf(lon);
    float Np = A_WGS / sqrtf(1.0f - E2 * sl * sl);
    float x = (Np + elev) * cl * co;
    float y = (Np + elev) * cl * so;
    float z = (Np * (1.0f - E2) + elev) * sl;

    const float NORM = 6500000.0f;
    float ux = clamp01((x / NORM + 1.0f) * 0.5f);
    float uy = clamp01((y / NORM + 1.0f) * 0.5f);
    float uz = clamp01((z / NORM + 1.0f) * 0.5f);
    float ut = clamp01(((t * 2.0f - 1.0f) * 0.9f + 1.0f) * 0.5f);

    float* outRow = out + (size_t)i * 192;            // [N, 4*24*2]
    encode_grid(ux, uy, uz, t_spatial, outRow + 0);   // spatial
    encode_grid(ux, uy, ut, t_xyt,     outRow + 48);  // xyt
    encode_grid(uy, uz, ut, t_yzt,     outRow + 96);  // yzt
    encode_grid(ux, uz, ut, t_xzt,     outRow + 144); // xzt
}

extern "C" void kernel_launch(void* const* d_in, const int* in_sizes, int n_in,
                              void* d_out, int out_size, void* d_ws, size_t ws_size,
                              hipStream_t stream) {
    (void)n_in; (void)d_ws; (void)ws_size; (void)out_size;
    const v4f*   coords    = (const v4f*)d_in[0];
    const float* t_spatial = (const float*)d_in[1];
    const float* t_xyt     = (const float*)d_in[2];
    const float* t_yzt     = (const float*)d_in[3];
    const float* t_xzt     = (const float*)d_in[4];
    float* out = (float*)d_out;

    int n = in_sizes[0] / 4;                   // coords is [N,4]
    const int threads = 256;                   // 8 wave32 waves per block
    int blocks = (n + threads - 1) / threads;
    hipLaunchKernelGGL(earth4d_kernel, dim3(blocks), dim3(threads), 0, stream,
                       coords, t_spatial, t_xyt, t_yzt, t_xzt, out, n);
}